// Horizontal_Convolutional_Layer_69861938037654
// MI455X (gfx1250) — compile-verified
//
#include <hip/hip_runtime.h>

typedef __attribute__((ext_vector_type(2))) float v2f;
typedef __attribute__((ext_vector_type(4))) float v4f;
typedef __attribute__((ext_vector_type(8))) float v8f;

#define SEQ 50
#define EMB 64
#define XSTRIDE 68        // padded LDS row stride for staged X (conflict-free A reads)
#define PSTRIDE 17        // padded LDS row stride for P tile
#define WAVES_PER_BLOCK 4
#define LDS_PER_WAVE (SEQ * XSTRIDE)   // 3400 words; P (64*17=1088) overlays the bottom

__global__ __launch_bounds__(WAVES_PER_BLOCK * 32) void hconv_wmma_kernel(
    const float* __restrict__ x,
    const float* __restrict__ w1, const float* __restrict__ b1,
    const float* __restrict__ w2, const float* __restrict__ b2,
    const float* __restrict__ w3, const float* __restrict__ b3,
    const float* __restrict__ w4, const float* __restrict__ b4,
    float* __restrict__ out, int B)
{
    __shared__ float lds[WAVES_PER_BLOCK][LDS_PER_WAVE];

    const int lane  = threadIdx.x & 31;
    const int wave  = threadIdx.x >> 5;
    const int batch = blockIdx.x * WAVES_PER_BLOCK + wave;
    if (batch >= B) return;

    const int n     = lane & 15;        // GEMM column = (branch, offset) pair
    const int khalf = (lane >> 4) * 2;  // K base selected by lane half

    // ---- B fragments (weights): built once, live in 32 VGPRs ----
    const float* wp = nullptr;
    if      (n == 0) wp = w1;
    else if (n <= 2) wp = w2 + (n - 1) * EMB;
    else if (n <= 5) wp = w3 + (n - 3) * EMB;
    else if (n <= 9) wp = w4 + (n - 6) * EMB;

    v2f bf[16];
#pragma unroll
    for (int k = 0; k < 16; ++k) {
        v2f t; t.x = 0.f; t.y = 0.f;
        if (wp) {
            const float* p = wp + khalf + 4 * k;  // K = khalf + {0,1} + 4k
            t.x = p[0];
            t.y = p[1];
        }
        bf[k] = t;
    }

    float*       L  = lds[wave];
    const float* xb = x + (size_t)batch * (SEQ * EMB);

    // ---- Phase 1: stage X into LDS with coalesced b128 loads ----
    // flat element p = i*128 + lane*4  ->  row = 2i + lane/16, col = (lane&15)*4
    {
        const float* src  = xb + lane * 4;
        const int    drow = lane >> 4;
        const int    dcol = (lane & 15) * 4;
#pragma unroll
        for (int i = 0; i < (SEQ * EMB) / 128; ++i) {           // 25 iters
            v4f v = *(const v4f*)(src + i * 128);
            *(v4f*)(L + (2 * i + drow) * XSTRIDE + dcol) = v;
        }
    }

    // ---- Phase 2: GEMM P[s,n] = sum_e X[s,e]*W[n,e] via V_WMMA_F32_16X16X4_F32 ----
#pragma unroll
    for (int t = 0; t < 4; ++t) {
        const int  row   = t * 16 + n;        // A-matrix M index == s
        const bool valid = (row < SEQ);       // zero-pad rows 50..63 (stay in-region)
        const float* ar  = L + row * XSTRIDE + khalf;

        v8f c = {0.f, 0.f, 0.f, 0.f, 0.f, 0.f, 0.f, 0.f};
#pragma unroll
        for (int k = 0; k < 16; ++k) {
            v2f a; a.x = 0.f; a.y = 0.f;
            if (valid) a = *(const v2f*)(ar + 4 * k);   // ds_load_b64, conflict-free
            c = __builtin_amdgcn_wmma_f32_16x16x4_f32(
                    false, a, false, bf[k], (short)0, c, false, false);
        }

        // C/D layout: s = 16t + v + 8*(lane>>4), col = n.
        // P overlay: max word 63*17+15 = 1086 < 16*68 = 1088, and tile t's stores
        // (words <= (16t+16)*17) never touch X rows >= 16(t+1) still to be read.
        const int sbase = t * 16 + (lane >> 4) * 8;
#pragma unroll
        for (int v = 0; v < 8; ++v)
            L[(sbase + v) * PSTRIDE + n] = c[v];
    }

    // ---- Phase 3: shifted-window sums + bias, ReLU, max over s ----
    const float bb1 = b1[0], bb2 = b2[0], bb3 = b3[0], bb4 = b4[0];
    float m1 = 0.f, m2 = 0.f, m3 = 0.f, m4 = 0.f;  // ReLU => max starts at 0

#pragma unroll
    for (int pass = 0; pass < 2; ++pass) {
        const int s = lane + pass * 32;
        if (s < SEQ) {
            const float* r0 = L + s * PSTRIDE;
            m1 = fmaxf(m1, r0[0] + bb1);
            if (s <= SEQ - 2)
                m2 = fmaxf(m2, r0[1] + r0[PSTRIDE + 2] + bb2);
            if (s <= SEQ - 3)
                m3 = fmaxf(m3, r0[3] + r0[PSTRIDE + 4] + r0[2 * PSTRIDE + 5] + bb3);
            if (s <= SEQ - 4)
                m4 = fmaxf(m4, r0[6] + r0[PSTRIDE + 7] + r0[2 * PSTRIDE + 8]
                               + r0[3 * PSTRIDE + 9] + bb4);
        }
    }

    // wave32 max reductions
#pragma unroll
    for (int off = 16; off > 0; off >>= 1) {
        m1 = fmaxf(m1, __shfl_xor(m1, off, 32));
        m2 = fmaxf(m2, __shfl_xor(m2, off, 32));
        m3 = fmaxf(m3, __shfl_xor(m3, off, 32));
        m4 = fmaxf(m4, __shfl_xor(m4, off, 32));
    }

    if (lane == 0) {
        float4 o; o.x = m1; o.y = m2; o.z = m3; o.w = m4;
        *(float4*)(out + (size_t)batch * 4) = o;
    }
}

extern "C" void kernel_launch(void* const* d_in, const int* in_sizes, int n_in,
                              void* d_out, int out_size, void* d_ws, size_t ws_size,
                              hipStream_t stream) {
    const float* x  = (const float*)d_in[0];
    const float* w1 = (const float*)d_in[1];
    const float* b1 = (const float*)d_in[2];
    const float* w2 = (const float*)d_in[3];
    const float* b2 = (const float*)d_in[4];
    const float* w3 = (const float*)d_in[5];
    const float* b3 = (const float*)d_in[6];
    const float* w4 = (const float*)d_in[7];
    const float* b4 = (const float*)d_in[8];
    float* out = (float*)d_out;

    const int B = in_sizes[0] / (SEQ * EMB);  // 32768
    dim3 grid((B + WAVES_PER_BLOCK - 1) / WAVES_PER_BLOCK);
    dim3 block(WAVES_PER_BLOCK * 32);

    hipLaunchKernelGGL(hconv_wmma_kernel, grid, block, 0, stream,
                       x, w1, b1, w2, b2, w3, b3, w4, b4, out, B);
}